// ColorConsistencyMetric_70557722739343
// MI455X (gfx1250) — compile-verified
//
#include <hip/hip_runtime.h>

typedef __attribute__((ext_vector_type(2))) float v2f;
typedef __attribute__((ext_vector_type(8))) float v8f;

#define IMG_H 1024
#define IMG_W 1024
#define REG_H 256
#define REG_W 256
// 32 batches * 3 channels * 16 regions = 1536 regions, each 256x256 floats.

// Kernel 1: one workgroup per region. 256 threads = 8 waves (wave32).
// Bulk accumulation runs through V_WMMA_F32_16X16X4_F32 with A = ones:
// D = Ones(16x4) x B(4x16) + C accumulates each B element into every D row,
// i.e. total over all D elements = 16 * (sum of all data fed as B). Exact in fp32.
__global__ __launch_bounds__(256) void cc_region_means_kernel(
    const float* __restrict__ images, float* __restrict__ region_means) {
  const int rid  = blockIdx.x;      // 0..1535
  const int pair = rid >> 4;        // b*3 + c  (plane index)
  const int reg  = rid & 15;        // region within plane
  const int gy   = reg >> 2;
  const int gx   = reg & 3;
  const size_t base = (size_t)pair * ((size_t)IMG_H * IMG_W)
                    + (size_t)gy * REG_H * IMG_W
                    + (size_t)gx * REG_W;

  const int tid  = threadIdx.x;
  const int wave = tid >> 5;
  const int lane = tid & 31;

  v2f ones; ones[0] = 1.0f; ones[1] = 1.0f;
  v8f acc = {0.f, 0.f, 0.f, 0.f, 0.f, 0.f, 0.f, 0.f};

  // Each wave owns 32 consecutive rows of the 256x256 region.
  const float* wbase = images + base + (size_t)(wave * 32) * IMG_W;

  for (int r = 0; r < 32; ++r) {
    const float* row = wbase + (size_t)r * IMG_W;
    __builtin_prefetch(row + 4 * IMG_W, 0, 0);   // global_prefetch_b8, 4 rows ahead
    // 256 floats per row: 32 lanes * 2 float4 each, fully coalesced.
    float4 p0 = *(const float4*)(row + lane * 4);
    float4 p1 = *(const float4*)(row + 128 + lane * 4);
    v2f b;
    b[0] = p0.x; b[1] = p0.y;
    acc = __builtin_amdgcn_wmma_f32_16x16x4_f32(false, ones, false, b, (short)0, acc, false, false);
    b[0] = p0.z; b[1] = p0.w;
    acc = __builtin_amdgcn_wmma_f32_16x16x4_f32(false, ones, false, b, (short)0, acc, false, false);
    b[0] = p1.x; b[1] = p1.y;
    acc = __builtin_amdgcn_wmma_f32_16x16x4_f32(false, ones, false, b, (short)0, acc, false, false);
    b[0] = p1.z; b[1] = p1.w;
    acc = __builtin_amdgcn_wmma_f32_16x16x4_f32(false, ones, false, b, (short)0, acc, false, false);
  }

  // Sum the 8 accumulator VGPRs, then exact wave32 butterfly reduce.
  float s = acc[0] + acc[1] + acc[2] + acc[3] + acc[4] + acc[5] + acc[6] + acc[7];
#pragma unroll
  for (int off = 16; off > 0; off >>= 1) s += __shfl_xor(s, off, 32);

  __shared__ float wsum[8];
  if (lane == 0) wsum[wave] = s;
  __syncthreads();
  if (tid == 0) {
    float t = 0.f;
#pragma unroll
    for (int i = 0; i < 8; ++i) t += wsum[i];
    // divide out the x16 WMMA row replication and the 65536 elements per region
    region_means[rid] = t * (1.0f / 16.0f) * (1.0f / 65536.0f);
  }
}

// Kernel 2: final scalar. region_means laid out as [pair(b*3+c)][region(16)].
__global__ __launch_bounds__(128) void cc_finalize_kernel(
    const float* __restrict__ region_means, float* __restrict__ out) {
  __shared__ float stds[96];
  const int t = threadIdx.x;

  if (t < 96) {
    float v[16];
    float m = 0.f;
#pragma unroll
    for (int i = 0; i < 16; ++i) { v[i] = region_means[t * 16 + i]; m += v[i]; }
    m *= (1.0f / 16.0f);
    float var = 0.f;
#pragma unroll
    for (int i = 0; i < 16; ++i) { float d = v[i] - m; var += d * d; }
    var *= (1.0f / 15.0f);               // ddof = 1 (unbiased)
    stds[t] = sqrtf(var);
  }
  __syncthreads();

  if (t < 32) {                           // one full wave: 32 batches
    float s = (stds[t * 3 + 0] + stds[t * 3 + 1] + stds[t * 3 + 2]) * (1.0f / 3.0f);
    float cons = 1.0f / (1.0f + s);
#pragma unroll
    for (int off = 16; off > 0; off >>= 1) cons += __shfl_xor(cons, off, 32);
    if (t == 0) out[0] = cons * (1.0f / 32.0f);
  }
}

extern "C" void kernel_launch(void* const* d_in, const int* in_sizes, int n_in,
                              void* d_out, int out_size, void* d_ws, size_t ws_size,
                              hipStream_t stream) {
  (void)in_sizes; (void)n_in; (void)out_size; (void)ws_size;
  const float* images = (const float*)d_in[0];
  float* out = (float*)d_out;
  float* ws  = (float*)d_ws;   // 1536 floats of scratch (region means)

  cc_region_means_kernel<<<1536, 256, 0, stream>>>(images, ws);
  cc_finalize_kernel<<<1, 128, 0, stream>>>(ws, out);
}